// AtomMPNN_26534307954800
// MI455X (gfx1250) — compile-verified
//
#include <hip/hip_runtime.h>
#include <hip/hip_bf16.h>
#include <math.h>

// Problem constants (match reference)
#define BB 4
#define NN 12288
#define KK 16
#define DD 128
#define IN1 257      // 2D+1
#define IN1P 260     // padded K-dim for GEMM1 (multiple of 4)
#define NBLK 96      // NN / 128
#define WAVES 4      // waves (nodes) per block in the message kernel

typedef float v2f __attribute__((ext_vector_type(2)));
typedef float v8f __attribute__((ext_vector_type(8)));

__device__ __forceinline__ float gelu_exact(float x) {
    return 0.5f * x * (1.0f + erff(x * 0.70710678118654752f));
}

// ---------------------------------------------------------------------------
// Kernel 0: transpose weights. w1t: (n=128, k=260 padded), w2t: (n=128, k=128)
// ---------------------------------------------------------------------------
__global__ __launch_bounds__(256) void k_transpose(
    const float* __restrict__ w1, const float* __restrict__ w2,
    float* __restrict__ w1t, float* __restrict__ w2t)
{
    int i = blockIdx.x * 256 + threadIdx.x;
    if (i < DD * IN1P) {
        int n = i / IN1P, k = i - n * IN1P;
        w1t[i] = (k < IN1) ? w1[k * DD + n] : 0.0f;
    }
    if (i < DD * DD) {
        int n = i / DD, k = i - n * DD;
        w2t[i] = w2[k * DD + n];
    }
}

// ---------------------------------------------------------------------------
// Kernel 1: per-node message MLP via fp32 WMMA, aggregate, residual -> updbuf
// One wave (32 lanes) per node. 16 neighbors = 16 WMMA rows.
// Branch-free, software-pipelined A-fragment path.
// ---------------------------------------------------------------------------
__global__ __launch_bounds__(WAVES * 32) void k_message(
    const float* __restrict__ emb,   const float* __restrict__ dists,
    const int*   __restrict__ eidx,  const float* __restrict__ amask,
    const float* __restrict__ w1t,   const float* __restrict__ b1,
    const float* __restrict__ w2t,   const float* __restrict__ b2,
    float* __restrict__ updbuf)
{
    __shared__ float h1s[WAVES][16 * 130];   // padded stride 130: conflict-free reads

    const int lane = threadIdx.x & 31;
    const int wave = threadIdx.x >> 5;
    const int row  = lane & 15;      // M row (neighbor slot) / N column within tile
    const int hi   = lane >> 4;      // K sub-offset selector (+2)
    const int node = blockIdx.x * WAVES + wave;
    const int b    = node / NN;
    const int n    = node - b * NN;

    // per-lane neighbor (row = lane&15, duplicated across halves)
    const int   e     = eidx[(size_t)(b * NN + n) * KK + row];
    const int   sidx  = (e < 0) ? 0 : e;
    const float* srcp = emb + ((size_t)b * NN + sidx) * DD + 2 * hi;
    const float smask = amask[b * NN + sidx];
    const float* selfp= emb + ((size_t)b * NN + n) * DD + 2 * hi;
    const float self_mask = amask[b * NN + n];
    const float dist  = dists[(size_t)(b * NN + n) * KK + row];

    unsigned long long bal = __ballot(e >= 0);
    const unsigned vb = (unsigned)(bal & 0xFFFFull);   // 16-bit valid mask (rows)
    const int nv = __popc(vb);
    const float inv_nv = 1.0f / (float)(nv == 0 ? 1 : nv);

    v8f acc[8];

    // 8 j-tile WMMAs sharing one A fragment; B fragments are contiguous v2f.
    auto wmma8 = [&](v2f a, const float* wp, int rowStride) {
        #pragma unroll
        for (int j = 0; j < 8; ++j) {
            v2f bb = *(const v2f*)(wp + (size_t)(16 * j) * rowStride);
            acc[j] = __builtin_amdgcn_wmma_f32_16x16x4_f32(
                false, a, false, bb, (short)0, acc[j], false, false);
        }
    };

    // ---------------- GEMM1: (16 x 260) @ (260 x 128), C init = b1 ----------
    #pragma unroll
    for (int j = 0; j < 8; ++j) {
        float bv = b1[16 * j + row];
        #pragma unroll
        for (int r = 0; r < 8; ++r) acc[j][r] = bv;
    }

    const float* w1base = w1t + (size_t)row * IN1P + 2 * hi;
    const float* w2base = w2t + (size_t)row * DD   + 2 * hi;

    // Phase 1: src neighbor features, cols 0..127 (32 K-steps), pipelined A.
    v2f araw = *(const v2f*)(srcp);
    #pragma unroll 1
    for (int t = 0; t < 32; ++t) {
        v2f a; a[0] = araw[0] * smask; a[1] = araw[1] * smask;
        // branch-free prefetch: next src fragment, or first self fragment
        const float* nap = (t < 31) ? (srcp + 4 * (t + 1)) : selfp;
        araw = *(const v2f*)(nap);
        wmma8(a, w1base + 4 * t, IN1P);
    }
    // Phase 2: self features, cols 128..255 (32 K-steps), pipelined A.
    #pragma unroll 1
    for (int t = 32; t < 64; ++t) {
        v2f a; a[0] = araw[0] * self_mask; a[1] = araw[1] * self_mask;
        const float* nap = (t < 63) ? (selfp + 4 * (t - 31)) : selfp; // dummy tail
        araw = *(const v2f*)(nap);
        wmma8(a, w1base + 4 * t, IN1P);
    }
    // Phase 3: distance column 256 (pad 257..259 are zero in w1t).
    {
        v2f a; a[0] = hi ? 0.0f : dist; a[1] = 0.0f;
        wmma8(a, w1base + 256, IN1P);
    }

    // GELU -> h1 into LDS in row-major (m, c) with padded stride
    #pragma unroll
    for (int j = 0; j < 8; ++j) {
        #pragma unroll
        for (int r = 0; r < 8; ++r) {
            const int m = r + 8 * hi;                // D-layout row
            h1s[wave][m * 130 + 16 * j + row] = gelu_exact(acc[j][r]);
        }
    }

    // ---------------- GEMM2: (16 x 128) @ (128 x 128), C init = b2 ----------
    #pragma unroll
    for (int j = 0; j < 8; ++j) {
        float bv = b2[16 * j + row];
        #pragma unroll
        for (int r = 0; r < 8; ++r) acc[j][r] = bv;
    }

    const float* h1p = &h1s[wave][row * 130 + 2 * hi];   // 8B-aligned
    v2f a2 = *(const v2f*)(h1p);
    #pragma unroll 1
    for (int t = 0; t < 32; ++t) {
        v2f a = a2;
        const int tn = (t < 31) ? (t + 1) : 0;           // dummy tail, branch-free
        a2 = *(const v2f*)(h1p + 4 * tn);
        wmma8(a, w2base + 4 * t, DD);
    }

    // GELU, mask invalid neighbors (per D-layout row), sum rows, residual, store
    #pragma unroll
    for (int j = 0; j < 8; ++j) {
        float p = 0.0f;
        #pragma unroll
        for (int r = 0; r < 8; ++r) {
            const int m = r + 8 * hi;
            const float vf = ((vb >> m) & 1u) ? 1.0f : 0.0f;
            p += gelu_exact(acc[j][r]) * vf;
        }
        p += __shfl_xor(p, 16, 32);                  // combine the two row-halves
        if (hi == 0) {
            const int ncol = 16 * j + row;
            const size_t o = ((size_t)(b * NN + n)) * DD + ncol;
            updbuf[o] = (emb[o] + p * inv_nv) * self_mask;
        }
    }
}

// ---------------------------------------------------------------------------
// Kernel 2: per-(b, 128-node block) partial sums / sumsq / mask count
// ---------------------------------------------------------------------------
__global__ __launch_bounds__(128) void k_partial(
    const float* __restrict__ updbuf, const float* __restrict__ amask,
    float* __restrict__ psum, float* __restrict__ psq, float* __restrict__ pcnt)
{
    const int blk = blockIdx.x;
    const int b   = blk / NBLK;
    const int nb  = blk - b * NBLK;
    const int d   = threadIdx.x;
    const float* base = updbuf + ((size_t)b * NN + (size_t)nb * 128) * DD + d;
    float s = 0.0f, q = 0.0f;
    #pragma unroll 4
    for (int i = 0; i < 128; ++i) {
        float v = base[(size_t)i * DD];
        s += v; q += v * v;
    }
    psum[(size_t)blk * DD + d] = s;
    psq [(size_t)blk * DD + d] = q;

    __shared__ float cs[128];
    cs[d] = amask[b * NN + nb * 128 + d];
    __syncthreads();
    for (int off = 64; off > 0; off >>= 1) {
        if (d < off) cs[d] += cs[d + off];
        __syncthreads();
    }
    if (d == 0) pcnt[blk] = cs[0];
}

// ---------------------------------------------------------------------------
// Kernel 3: finalize per-(b,d) mean / rstd (deterministic tree, no atomics)
// ---------------------------------------------------------------------------
__global__ __launch_bounds__(128) void k_stats(
    const float* __restrict__ psum, const float* __restrict__ psq,
    const float* __restrict__ pcnt,
    float* __restrict__ meanbuf, float* __restrict__ rstdbuf)
{
    const int b = blockIdx.x;
    const int d = threadIdx.x;
    float s = 0.0f, q = 0.0f, c = 0.0f;
    for (int i = 0; i < NBLK; ++i) {
        s += psum[((size_t)b * NBLK + i) * DD + d];
        q += psq [((size_t)b * NBLK + i) * DD + d];
        c += pcnt[b * NBLK + i];
    }
    if (c == 0.0f) c = 1.0f;
    const float mean = s / c;
    float var = q / c - mean * mean;
    if (var < 0.0f) var = 0.0f;
    meanbuf[b * DD + d] = mean;
    rstdbuf[b * DD + d] = rsqrtf(var + 1e-5f);
}

// ---------------------------------------------------------------------------
// Kernel 4: normalize + affine + mask -> out
// ---------------------------------------------------------------------------
__global__ __launch_bounds__(256) void k_norm(
    const float* __restrict__ updbuf, const float* __restrict__ amask,
    const float* __restrict__ meanbuf, const float* __restrict__ rstdbuf,
    const float* __restrict__ gamma, const float* __restrict__ beta,
    float* __restrict__ out)
{
    const size_t i = (size_t)blockIdx.x * 256 + threadIdx.x;
    const int d = (int)(i & (DD - 1));
    const size_t bn = i >> 7;
    const int b = (int)(bn / NN);
    const float m = amask[bn];
    const float v = updbuf[i];
    out[i] = ((v - meanbuf[b * DD + d]) * rstdbuf[b * DD + d] * gamma[d] + beta[d]) * m;
}

// ---------------------------------------------------------------------------
extern "C" void kernel_launch(void* const* d_in, const int* in_sizes, int n_in,
                              void* d_out, int out_size, void* d_ws, size_t ws_size,
                              hipStream_t stream) {
    const float* emb   = (const float*)d_in[0];
    const float* dists = (const float*)d_in[1];
    const int*   eidx  = (const int*)  d_in[2];
    const float* amask = (const float*)d_in[3];
    const float* w1    = (const float*)d_in[4];
    const float* b1    = (const float*)d_in[5];
    const float* w2    = (const float*)d_in[6];
    const float* b2    = (const float*)d_in[7];
    const float* gamma = (const float*)d_in[8];
    const float* beta  = (const float*)d_in[9];
    float* out = (float*)d_out;

    // workspace carve-up (floats)
    char* ws = (char*)d_ws;
    float* updbuf = (float*)ws;                     ws += (size_t)BB * NN * DD * sizeof(float);
    float* w1t    = (float*)ws;                     ws += (size_t)DD * IN1P   * sizeof(float);
    float* w2t    = (float*)ws;                     ws += (size_t)DD * DD     * sizeof(float);
    float* psum   = (float*)ws;                     ws += (size_t)BB * NBLK * DD * sizeof(float);
    float* psq    = (float*)ws;                     ws += (size_t)BB * NBLK * DD * sizeof(float);
    float* pcnt   = (float*)ws;                     ws += (size_t)BB * NBLK * sizeof(float);
    float* meanb  = (float*)ws;                     ws += (size_t)BB * DD * sizeof(float);
    float* rstdb  = (float*)ws;

    k_transpose<<<(DD * IN1P + 255) / 256, 256, 0, stream>>>(w1, w2, w1t, w2t);

    const int nodes = BB * NN;
    k_message<<<nodes / WAVES, WAVES * 32, 0, stream>>>(
        emb, dists, eidx, amask, w1t, b1, w2t, b2, updbuf);

    k_partial<<<BB * NBLK, 128, 0, stream>>>(updbuf, amask, psum, psq, pcnt);
    k_stats<<<BB, 128, 0, stream>>>(psum, psq, pcnt, meanb, rstdb);
    k_norm<<<(int)(((size_t)BB * NN * DD) / 256), 256, 0, stream>>>(
        updbuf, amask, meanb, rstdb, gamma, beta, out);
}